// DotGATModel_42863773614472
// MI455X (gfx1250) — compile-verified
//
#include <hip/hip_runtime.h>

#define D 128               // feature dim (D_IN == D_H == 128 in the reference)

typedef __attribute__((ext_vector_type(2))) float v2f;
typedef __attribute__((ext_vector_type(8))) float v8f;

// ---------------------------------------------------------------------------
// GEMM: Out[n, j] = sum_k act(X[n, k]) * W[j, k]   (Out = act(X) @ W^T)
// One wave computes a 32x16 output block (two 16x16 WMMA tiles sharing the
// B fetch).  V_WMMA_F32_16X16X4_F32, K stepped by 4.  Block = 256 threads =
// 8 waves = the 8 column tiles of one 32-row stripe.
// RELU is a template parameter so the layer-1 inner loop is pure load->wmma.
// ---------------------------------------------------------------------------
template<bool RELU>
__global__ __launch_bounds__(256)
void gemm_xwt_wmma(const float* __restrict__ X, const float* __restrict__ W,
                   float* __restrict__ Out, int n_rows) {
    const int lane = threadIdx.x & 31;
    const int wv   = threadIdx.x >> 5;         // column tile 0..7
    const int row0 = blockIdx.x << 5;          // 32 rows per block
    const int col0 = wv << 4;
    const int m = lane & 15;                   // row-within-tile (A) / col (B,D)
    const int g = lane >> 4;                   // half-wave selector

    int ar0 = row0 + m;       if (ar0 >= n_rows) ar0 = n_rows - 1;
    int ar1 = row0 + 16 + m;  if (ar1 >= n_rows) ar1 = n_rows - 1;
    const float* a0p = X + (size_t)ar0 * D + 2 * g;           // A0[m][2g..2g+1]
    const float* a1p = X + (size_t)ar1 * D + 2 * g;           // A1[m][2g..2g+1]
    const float* bp  = W + (size_t)(col0 + m) * D + 2 * g;    // B[2g][n]=W[col0+n][k]

    v8f acc0 = {};
    v8f acc1 = {};
    #pragma unroll 4
    for (int kk = 0; kk < D; kk += 4) {
        v2f b  = *(const v2f*)(bp  + kk);
        v2f a0 = *(const v2f*)(a0p + kk);
        v2f a1 = *(const v2f*)(a1p + kk);
        if (RELU) {
            a0[0] = __builtin_fmaxf(a0[0], 0.0f); a0[1] = __builtin_fmaxf(a0[1], 0.0f);
            a1[0] = __builtin_fmaxf(a1[0], 0.0f); a1[1] = __builtin_fmaxf(a1[1], 0.0f);
        }
        acc0 = __builtin_amdgcn_wmma_f32_16x16x4_f32(false, a0, false, b,
                                                     (short)0, acc0, false, false);
        acc1 = __builtin_amdgcn_wmma_f32_16x16x4_f32(false, a1, false, b,
                                                     (short)0, acc1, false, false);
    }

    // D layout: vgpr r -> (row = r + 8g, col = m)
    if (row0 + 32 <= n_rows) {
        // fast path: branchless stores off one base pointer, immediate offsets
        float* q0 = Out + (size_t)(row0 + 8 * g) * D + col0 + m;
        float* q1 = q0 + (size_t)16 * D;
        #pragma unroll
        for (int r = 0; r < 8; ++r) {
            q0[(size_t)r * D] = acc0[r];
            q1[(size_t)r * D] = acc1[r];
        }
    } else {
        #pragma unroll
        for (int r = 0; r < 8; ++r) {
            int row = row0 + r + 8 * g;
            if (row < n_rows)      Out[(size_t)row * D + col0 + m] = acc0[r];
            if (row + 16 < n_rows) Out[(size_t)(row + 16) * D + col0 + m] = acc1[r];
        }
    }
}

// ---------------------------------------------------------------------------
// Per-layer node-state init: max = -inf, denom = 0, accumulation rows = 0
// ---------------------------------------------------------------------------
__global__ void init_nodes(float* __restrict__ mmax, float* __restrict__ denom,
                           float* __restrict__ acc, int n) {
    int i = blockIdx.x * blockDim.x + threadIdx.x;
    if (i < n * D) acc[i] = 0.0f;
    if (i < n) { mmax[i] = -__builtin_inff(); denom[i] = 0.0f; }
}

// ---------------------------------------------------------------------------
// Edge dot: a[e] = (ft[dst] . ft[src]) / sqrt(D).  One wave32 per edge.
// ---------------------------------------------------------------------------
__global__ __launch_bounds__(256)
void edge_dot(const float* __restrict__ ft, const int* __restrict__ src,
              const int* __restrict__ dst, float* __restrict__ a_out, int E) {
    int e = blockIdx.x * (blockDim.x >> 5) + (threadIdx.x >> 5);
    int lane = threadIdx.x & 31;
    if (e >= E) return;
    int s = src[e], d = dst[e];
    const float4* ps = (const float4*)(ft + (size_t)s * D);
    const float4* pd = (const float4*)(ft + (size_t)d * D);
    float4 vs = ps[lane];
    float4 vd = pd[lane];
    float dot = vs.x * vd.x + vs.y * vd.y + vs.z * vd.z + vs.w * vd.w;
    #pragma unroll
    for (int off = 16; off; off >>= 1) dot += __shfl_xor(dot, off, 32);
    if (lane == 0) a_out[e] = dot * 0.08838834764831845f;   // 1/sqrt(128)
}

// ---------------------------------------------------------------------------
// Segment max over incoming edges (ordered-int-bits float atomic max trick)
// ---------------------------------------------------------------------------
__device__ __forceinline__ void atomicMaxF(float* addr, float val) {
    if (val >= 0.0f) atomicMax((int*)addr, __float_as_int(val));
    else             atomicMin((unsigned int*)addr, __float_as_uint(val));
}

__global__ void edge_max(const float* __restrict__ a, const int* __restrict__ dst,
                         float* __restrict__ mmax, int E) {
    int e = blockIdx.x * blockDim.x + threadIdx.x;
    if (e >= E) return;
    atomicMaxF(&mmax[dst[e]], a[e]);
}

// ---------------------------------------------------------------------------
// ex[e] = exp(a[e] - m[dst]) (in place), denom[dst] += ex[e]
// ---------------------------------------------------------------------------
__global__ void edge_exp(float* __restrict__ a, const int* __restrict__ dst,
                         const float* __restrict__ mmax, float* __restrict__ denom,
                         int E) {
    int e = blockIdx.x * blockDim.x + threadIdx.x;
    if (e >= E) return;
    int d = dst[e];
    float v = __expf(a[e] - mmax[d]);
    a[e] = v;
    atomicAdd(&denom[d], v);
}

// ---------------------------------------------------------------------------
// out[dst] += ft[src] * ex[e]/denom[dst].  One wave32 per edge, f32 atomics.
// ---------------------------------------------------------------------------
__global__ __launch_bounds__(256)
void edge_scatter(const float* __restrict__ ft, const int* __restrict__ src,
                  const int* __restrict__ dst, const float* __restrict__ ex,
                  const float* __restrict__ denom, float* __restrict__ out, int E) {
    int e = blockIdx.x * (blockDim.x >> 5) + (threadIdx.x >> 5);
    int lane = threadIdx.x & 31;
    if (e >= E) return;
    int s = src[e], d = dst[e];
    float w = ex[e] / denom[d];
    const float4* ps = (const float4*)(ft + (size_t)s * D);
    float4 v = ps[lane];
    float* q = out + (size_t)d * D + lane * 4;
    atomicAdd(q + 0, v.x * w);
    atomicAdd(q + 1, v.y * w);
    atomicAdd(q + 2, v.z * w);
    atomicAdd(q + 3, v.w * w);
}

// ---------------------------------------------------------------------------
// Pooling + classifier
// ---------------------------------------------------------------------------
__global__ void zero_pool(float* __restrict__ hgsum, float* __restrict__ counts, int B) {
    int i = blockIdx.x * blockDim.x + threadIdx.x;
    if (i < B * D) hgsum[i] = 0.0f;
    if (i < B) counts[i] = 0.0f;
}

__global__ void pool_count(const int* __restrict__ gid, float* __restrict__ counts, int N) {
    int n = blockIdx.x * blockDim.x + threadIdx.x;
    if (n < N) atomicAdd(&counts[gid[n]], 1.0f);
}

__global__ __launch_bounds__(256)
void pool_sum(const float* __restrict__ h2, const int* __restrict__ gid,
              float* __restrict__ hgsum, int N) {
    int n = blockIdx.x * (blockDim.x >> 5) + (threadIdx.x >> 5);
    int lane = threadIdx.x & 31;
    if (n >= N) return;
    int b = gid[n];
    const float4* p = (const float4*)(h2 + (size_t)n * D);
    float4 v = p[lane];
    float* q = hgsum + (size_t)b * D + lane * 4;
    atomicAdd(q + 0, v.x);
    atomicAdd(q + 1, v.y);
    atomicAdd(q + 2, v.z);
    atomicAdd(q + 3, v.w);
}

__global__ void classifier(const float* __restrict__ hgsum, const float* __restrict__ counts,
                           const float* __restrict__ perm, const float* __restrict__ Wc,
                           const float* __restrict__ bc, float* __restrict__ out,
                           int B, int NC, int PERM_LEN) {
    int t = blockIdx.x * blockDim.x + threadIdx.x;
    if (t >= B * NC) return;
    int b = t / NC, c = t % NC;
    float inv = 1.0f / __builtin_fmaxf(counts[b], 1.0f);
    const float* wrow = Wc + (size_t)c * (D + PERM_LEN);
    float s = bc[c];
    for (int j = 0; j < D; ++j)        s += hgsum[(size_t)b * D + j] * inv * wrow[j];
    for (int p = 0; p < PERM_LEN; ++p) s += perm[(size_t)b * PERM_LEN + p] * wrow[D + p];
    out[(size_t)b * NC + c] = s;
}

// ---------------------------------------------------------------------------
extern "C" void kernel_launch(void* const* d_in, const int* in_sizes, int n_in,
                              void* d_out, int out_size, void* d_ws, size_t ws_size,
                              hipStream_t stream) {
    const float* h    = (const float*)d_in[0];
    const float* perm = (const float*)d_in[1];
    const float* W1   = (const float*)d_in[2];
    const float* W2   = (const float*)d_in[3];
    const float* Wc   = (const float*)d_in[4];
    const float* bc   = (const float*)d_in[5];
    const int*   src  = (const int*)d_in[6];
    const int*   dst  = (const int*)d_in[7];
    const int*   gid  = (const int*)d_in[8];

    const int N  = in_sizes[8];
    const int E  = in_sizes[6];
    const int NC = in_sizes[5];
    const int PERM_LEN = in_sizes[4] / NC - D;   // Wc row = D_H + PERM
    const int B  = in_sizes[1] / PERM_LEN;

    // Workspace layout (floats): ~44 MB total, fits easily in MI455X's 192MB L2
    float* ws = (float*)d_ws;
    size_t off = 0;
    float* ft     = ws + off; off += (size_t)N * D;   // per-layer transformed feats
    float* hbuf   = ws + off; off += (size_t)N * D;   // aggregated output (reused)
    float* mmax   = ws + off; off += (size_t)N;
    float* denom  = ws + off; off += (size_t)N;
    float* ea     = ws + off; off += (size_t)E;       // edge score -> edge exp
    float* hgsum  = ws + off; off += (size_t)B * D;
    float* counts = ws + off; off += (size_t)B;
    (void)ws_size; (void)n_in; (void)out_size;

    const dim3 blk(256);
    const int row_blocks   = (N + 31) / 32;   // 32 rows per GEMM block
    const int nd_blocks    = (int)(((size_t)N * D + 255) / 256);
    const int ewave_blocks = (E + 7) / 8;     // 8 waves / block, 1 wave / edge
    const int ethr_blocks  = (E + 255) / 256;
    const int nwave_blocks = (N + 7) / 8;

    // ---- Layer 1: ft1 = h @ W1^T ; attention ; h1 = scatter ----
    gemm_xwt_wmma<false><<<row_blocks, blk, 0, stream>>>(h, W1, ft, N);
    init_nodes<<<nd_blocks, blk, 0, stream>>>(mmax, denom, hbuf, N);
    edge_dot<<<ewave_blocks, blk, 0, stream>>>(ft, src, dst, ea, E);
    edge_max<<<ethr_blocks, blk, 0, stream>>>(ea, dst, mmax, E);
    edge_exp<<<ethr_blocks, blk, 0, stream>>>(ea, dst, mmax, denom, E);
    edge_scatter<<<ewave_blocks, blk, 0, stream>>>(ft, src, dst, ea, denom, hbuf, E);

    // ---- Layer 2: ft2 = relu(h1) @ W2^T (relu fused into GEMM A-read) ----
    gemm_xwt_wmma<true><<<row_blocks, blk, 0, stream>>>(hbuf, W2, ft, N);
    init_nodes<<<nd_blocks, blk, 0, stream>>>(mmax, denom, hbuf, N);  // hbuf -> h2 acc
    edge_dot<<<ewave_blocks, blk, 0, stream>>>(ft, src, dst, ea, E);
    edge_max<<<ethr_blocks, blk, 0, stream>>>(ea, dst, mmax, E);
    edge_exp<<<ethr_blocks, blk, 0, stream>>>(ea, dst, mmax, denom, E);
    edge_scatter<<<ewave_blocks, blk, 0, stream>>>(ft, src, dst, ea, denom, hbuf, E);

    // ---- Mean-pool per graph + classifier head ----
    zero_pool<<<(B * D + 255) / 256, blk, 0, stream>>>(hgsum, counts, B);
    pool_count<<<(N + 255) / 256, blk, 0, stream>>>(gid, counts, N);
    pool_sum<<<nwave_blocks, blk, 0, stream>>>(hbuf, gid, hgsum, N);
    classifier<<<1, 256, 0, stream>>>(hgsum, counts, perm, Wc, bc,
                                      (float*)d_out, B, NC, PERM_LEN);
}